// RareVariantAwareFusion_19662360281179
// MI455X (gfx1250) — compile-verified
//
#include <hip/hip_runtime.h>
#include <math.h>

typedef __attribute__((ext_vector_type(2))) float v2f;
typedef __attribute__((ext_vector_type(4))) float v4f;
typedef __attribute__((ext_vector_type(8))) float v8f;

#define B_   16
#define K_   16
#define L_   512
#define D_   256
#define H_   16
#define ROWS 32          // rows (b,l) per block
#define PSTR 261         // odd-padded LDS row stride -> conflict-free column access

// One k-step (k..k+3) for a wave's 16(M) x 64(N) strip: 4 WMMAs sharing one A fragment.
__device__ __forceinline__ void wmma4(const v2f a, const float* __restrict__ wfp,
                                      v8f& c0, v8f& c1, v8f& c2, v8f& c3) {
    v2f bm;
    bm.x = wfp[0];        bm.y = wfp[D_];
    c0 = __builtin_amdgcn_wmma_f32_16x16x4_f32(false, a, false, bm, (short)0, c0, false, false);
    bm.x = wfp[16];       bm.y = wfp[D_ + 16];
    c1 = __builtin_amdgcn_wmma_f32_16x16x4_f32(false, a, false, bm, (short)0, c1, false, false);
    bm.x = wfp[32];       bm.y = wfp[D_ + 32];
    c2 = __builtin_amdgcn_wmma_f32_16x16x4_f32(false, a, false, bm, (short)0, c2, false, false);
    bm.x = wfp[48];       bm.y = wfp[D_ + 48];
    c3 = __builtin_amdgcn_wmma_f32_16x16x4_f32(false, a, false, bm, (short)0, c3, false, false);
}

__global__ __launch_bounds__(256) void rare_variant_fusion_kernel(
    const float* __restrict__ orig,   // [B,L,D]
    const float* __restrict__ rag,    // [B,K,L,D]
    const float* __restrict__ af,     // [B,L]
    const float* __restrict__ w1,     // [1,H]
    const float* __restrict__ b1,     // [H]
    const float* __restrict__ w2,     // [H,D]
    const float* __restrict__ b2,     // [D]
    const float* __restrict__ wf,     // [2D,D]
    const float* __restrict__ bf,     // [D]
    const float* __restrict__ ln_g,   // [D]
    const float* __restrict__ ln_b,   // [D]
    float* __restrict__ out)          // [B,L,D]
{
    __shared__ float xp[ROWS * PSTR];   // pooled (then reused as y) : 32 x 256 padded
    __shared__ float s_lds[ROWS];

    const int t  = threadIdx.x;
    const int m0 = blockIdx.x * ROWS;   // first flat row  (m = b*L + l)
    const int b  = m0 / L_;             // constant within block (L % ROWS == 0)
    const int l0 = m0 % L_;

    if (t < ROWS) s_lds[t] = 0.0f;

    // ---------------- Stage 1: K-reduction of rag_feat (HBM-bound bulk) ----------------
    // thread t: row-group t/64 (rows rg, rg+4, ..., rg+28), float4 column t%64.
    {
        const int rg = t >> 6;
        const int c4 = (t & 63) << 2;
        const size_t kstride = (size_t)L_ * D_;
        #pragma unroll
        for (int j = 0; j < 8; ++j) {
            const int r = rg + (j << 2);
            const float* base = rag + (((size_t)b * K_) * L_ + (size_t)(l0 + r)) * D_ + c4;
            v4f sum = {0.f, 0.f, 0.f, 0.f};
            v4f mx  = {-3.402823466e38f, -3.402823466e38f, -3.402823466e38f, -3.402823466e38f};
            #pragma unroll
            for (int k = 0; k < K_; ++k) {
                v4f v = __builtin_nontemporal_load((const v4f*)(base + (size_t)k * kstride));
                sum += v;
                mx.x = fmaxf(mx.x, v.x); mx.y = fmaxf(mx.y, v.y);
                mx.z = fmaxf(mx.z, v.z); mx.w = fmaxf(mx.w, v.w);
            }
            v4f pv = sum * (0.7f / (float)K_) + mx * 0.3f;   // 0.7*mean + 0.3*max
            float* prow = &xp[r * PSTR + c4];
            prow[0] = pv.x; prow[1] = pv.y; prow[2] = pv.z; prow[3] = pv.w;
        }
    }
    __syncthreads();

    // ---------------- Stage 2: s[r] = sum_d sigmoid(relu(af*w1+b1) @ w2 + b2) ----------
    // 8 threads per row, 32 d-columns each, LDS float atomics to combine.
    {
        const int r     = t >> 3;
        const int chunk = t & 7;
        const float a = af[m0 + r];
        float h[H_];
        #pragma unroll
        for (int j = 0; j < H_; ++j) h[j] = fmaxf(a * w1[j] + b1[j], 0.0f);
        float partial = 0.0f;
        for (int dd = 0; dd < 32; ++dd) {
            const int d = (chunk << 5) + dd;
            float acc = b2[d];
            #pragma unroll
            for (int j = 0; j < H_; ++j) acc += h[j] * w2[j * D_ + d];
            partial += 1.0f / (1.0f + expf(-acc));
        }
        atomicAdd(&s_lds[r], partial);   // ds_add_f32
    }
    __syncthreads();

    // ---------------- Stage 3: apply scalar gate to pooled ----------------
    #pragma unroll
    for (int j = 0; j < 32; ++j) {
        const int f = t + (j << 8);           // 0..8191 over 32x256
        const int r = f >> 8, c = f & 255;
        xp[r * PSTR + c] *= s_lds[r];
    }
    __syncthreads();

    // ---------------- Stage 4: GEMM y = [orig | pooled] @ wf via v_wmma_f32_16x16x4 ----
    const int lane  = t & 31;
    const int w     = t >> 5;          // wave 0..7
    const int mt    = w & 1;           // M tile: rows mt*16 .. mt*16+15
    const int nq    = w >> 1;          // N quad: cols nq*64 .. nq*64+63
    const int mrow  = lane & 15;       // A/B lane mapping per ISA 7.12.2
    const int khalf = (lane >> 4) << 1;
    const int ncol  = nq * 64 + (lane & 15);

    v8f c0 = {}, c1 = {}, c2 = {}, c3 = {};

    // first half of K-dim: A from global orig
    {
        const float* arow = orig + (size_t)(m0 + mt * 16 + mrow) * D_;
        for (int k = 0; k < D_; k += 4) {
            v2f a; a.x = arow[k + khalf]; a.y = arow[k + khalf + 1];
            wmma4(a, wf + (size_t)(k + khalf) * D_ + ncol, c0, c1, c2, c3);
        }
    }
    // second half of K-dim: A from LDS pooled (gated)
    {
        const float* prow = &xp[(mt * 16 + mrow) * PSTR];
        for (int k = D_; k < 2 * D_; k += 4) {
            v2f a; a.x = prow[k - D_ + khalf]; a.y = prow[k - D_ + khalf + 1];
            wmma4(a, wf + (size_t)(k + khalf) * D_ + ncol, c0, c1, c2, c3);
        }
    }
    __syncthreads();   // all waves done reading xp before y overwrites it

    // ---------------- Stage 5: spill C tiles (y) into LDS ----------------
    {
        const int row0 = mt * 16 + ((lane >> 4) << 3);   // C layout: VGPR i -> M=i(+8)
        const int col0 = nq * 64 + (lane & 15);
        #pragma unroll
        for (int vi = 0; vi < 8; ++vi) {
            float* yrow = &xp[(row0 + vi) * PSTR];
            yrow[col0 +  0] = c0[vi];
            yrow[col0 + 16] = c1[vi];
            yrow[col0 + 32] = c2[vi];
            yrow[col0 + 48] = c3[vi];
        }
    }
    __syncthreads();

    // ---------------- Stage 6: LayerNorm + GELU + gated residual ----------------
    // wave w handles rows 4w..4w+3; lane covers 8 consecutive d's.
    {
        const int dbase = lane << 3;
        for (int rr = 0; rr < 4; ++rr) {
            const int row = (w << 2) + rr;
            const float* yrow = &xp[row * PSTR];
            float yv[8];
            float sum = 0.0f, sq = 0.0f;
            #pragma unroll
            for (int j = 0; j < 8; ++j) {
                float y = yrow[dbase + j] + bf[dbase + j];
                yv[j] = y; sum += y; sq += y * y;
            }
            #pragma unroll
            for (int off = 16; off >= 1; off >>= 1) {
                sum += __shfl_xor(sum, off);
                sq  += __shfl_xor(sq,  off);
            }
            const float mu  = sum * (1.0f / (float)D_);
            const float var = sq * (1.0f / (float)D_) - mu * mu;
            const float rs  = rsqrtf(var + 1e-5f);
            const float a   = af[m0 + row];
            const float g   = sqrtf(a * (1.0f - a));
            const float* orow = orig + (size_t)(m0 + row) * D_ + dbase;
            float res[8];
            #pragma unroll
            for (int j = 0; j < 8; ++j) {
                const int d = dbase + j;
                const float yn = (yv[j] - mu) * rs * ln_g[d] + ln_b[d];
                const float ge = 0.5f * yn * (1.0f + erff(yn * 0.70710678118654752f));
                res[j] = orow[j] + ge * g;
            }
            v4f* o4 = (v4f*)(out + (size_t)(m0 + row) * D_ + dbase);
            v4f s0 = {res[0], res[1], res[2], res[3]};
            v4f s1 = {res[4], res[5], res[6], res[7]};
            o4[0] = s0; o4[1] = s1;
        }
    }
}

extern "C" void kernel_launch(void* const* d_in, const int* in_sizes, int n_in,
                              void* d_out, int out_size, void* d_ws, size_t ws_size,
                              hipStream_t stream) {
    const float* orig = (const float*)d_in[0];
    const float* rag  = (const float*)d_in[1];
    const float* af   = (const float*)d_in[2];
    const float* w1   = (const float*)d_in[3];
    const float* b1   = (const float*)d_in[4];
    const float* w2   = (const float*)d_in[5];
    const float* b2   = (const float*)d_in[6];
    const float* wf   = (const float*)d_in[7];
    const float* bf   = (const float*)d_in[8];
    const float* lng  = (const float*)d_in[9];
    const float* lnb  = (const float*)d_in[10];
    float* out = (float*)d_out;

    dim3 grid((B_ * L_) / ROWS);   // 256 blocks of 32 rows
    dim3 block(256);               // 8 wave32s
    rare_variant_fusion_kernel<<<grid, block, 0, stream>>>(
        orig, rag, af, w1, b1, w2, b2, wf, bf, lng, lnb, out);
}